// LightGCN_implicit_19688130085763
// MI455X (gfx1250) — compile-verified
//
#include <hip/hip_runtime.h>
#include <hip/hip_bf16.h>
#include <stdint.h>

#define U_NODES 100000
#define I_NODES 50000
#define N_NODES 150000
#define DIM     64
#define BATCH   4096
#define CHUNK   2048   // edges per workgroup, streamed by TDM

typedef __attribute__((ext_vector_type(4))) unsigned int v4u;
typedef __attribute__((ext_vector_type(8))) int          v8i;
typedef __attribute__((ext_vector_type(4))) int          v4i;

// ---- LDS byte-offset of a __shared__ object (generic -> AS3 -> int) ----
__device__ __forceinline__ unsigned lds_offset(const void* p) {
    return (unsigned)(size_t)(__attribute__((address_space(3))) const void*)p;
}

// ---- Guaranteed-native FP32 scatter-add (GLOBAL_ATOMIC_ADD_F32, no CAS) --
// Non-returning (TH=0 -> no pre-op value, STOREcnt-tracked), device scope so
// the RMW is performed at L2 where all WGPs are coherent.
__device__ __forceinline__ void gadd_f32(float* p, float v) {
    asm volatile("global_atomic_add_f32 %0, %1, off scope:SCOPE_DEV"
                 :: "v"((uint64_t)(uintptr_t)p), "v"(v)
                 : "memory");
}

// ---- Build a 1-row TDM descriptor and issue tensor_load_to_lds ----------
// Loads tileElems 4-byte elements from gptr into LDS at lds_off.
// tensorElems = #valid elements remaining at gptr; reads past it return 0
// (zero-fill handles the ragged tail chunk for free).
__device__ __forceinline__ void tdm_load_1d(unsigned lds_off, const void* gptr,
                                            unsigned tileElems, unsigned tensorElems) {
    uint64_t ga = (uint64_t)(uintptr_t)gptr;
    v4u g0;
    g0.x = 1u;                                   // count=1 (valid user descriptor)
    g0.y = lds_off;                              // lds_addr (bytes)
    g0.z = (unsigned)(ga & 0xFFFFFFFFu);         // global_addr[31:0]
    g0.w = (unsigned)((ga >> 32) & 0x1FFFFFFu)   // global_addr[56:32]
         | (2u << 30);                           // type = 2 ("image")
    v8i g1;
    g1[0] = (int)(2u << 16);                     // data_size=2 (4B); wg_mask=0
    g1[1] = (int)((tensorElems & 0xFFFFu) << 16);        // tensor_dim0[15:0]
    g1[2] = (int)(((tensorElems >> 16) & 0xFFFFu)        // tensor_dim0[31:16]
         |  (1u << 16));                                 // tensor_dim1 = 1
    g1[3] = (int)((tileElems & 0xFFFFu) << 16);          // tile_dim0
    g1[4] = 1;                                           // tile_dim1 = 1 (tile_dim2=0)
    g1[5] = (int)tileElems;                              // tensor_dim0_stride[31:0]
    g1[6] = 0;
    g1[7] = 0;
    v4i z4 = (v4i){0, 0, 0, 0};                  // groups 2/3 unused (<=2D tensor)
    v8i z8 = (v8i){0, 0, 0, 0, 0, 0, 0, 0};      // extra group (clang-23 6-arg form)
    __builtin_amdgcn_tensor_load_to_lds(g0, g1, z4, z4, z8, /*cpol=*/0);
}

// ---- cur0 = acc = concat(user_emb, item_emb) ---------------------------
__global__ __launch_bounds__(256) void lgcn_init(const float4* __restrict__ ue,
                                                 const float4* __restrict__ ie,
                                                 float4* __restrict__ cur,
                                                 float4* __restrict__ acc,
                                                 unsigned n4) {
    unsigned i = blockIdx.x * 256u + threadIdx.x;
    if (i >= n4) return;
    unsigned node = i >> 4;                      // 16 float4 per node (DIM=64)
    float4 e = (node < U_NODES) ? ue[i] : ie[i - (unsigned)U_NODES * 16u];
    cur[i] = e;
    acc[i] = e;
}

__global__ __launch_bounds__(256) void lgcn_zero(float4* __restrict__ y, unsigned n4) {
    unsigned i = blockIdx.x * 256u + threadIdx.x;
    if (i < n4) y[i] = make_float4(0.f, 0.f, 0.f, 0.f);
}

// ---- SpMM: y[row] += val * x[col], edges streamed to LDS by the TDM ----
__global__ __launch_bounds__(256) void lgcn_spmm(const float* __restrict__ val,
                                                 const int*   __restrict__ row,
                                                 const int*   __restrict__ col,
                                                 const float* __restrict__ x,
                                                 float*       __restrict__ y,
                                                 int nnz) {
    __shared__ float sval[CHUNK];
    __shared__ int   srow[CHUNK];
    __shared__ int   scol[CHUNK];

    const int chunkStart = (int)blockIdx.x * CHUNK;
    if (threadIdx.x == 0) {
        // Whole waves with EXEC==0 branch over this; the one wave holding
        // lane 0 issues the DMAs (TDM ignores EXEC) and waits for them.
        unsigned remaining = (unsigned)(nnz - chunkStart);
        tdm_load_1d(lds_offset(sval), val + chunkStart, CHUNK, remaining);
        tdm_load_1d(lds_offset(srow), row + chunkStart, CHUNK, remaining);
        tdm_load_1d(lds_offset(scol), col + chunkStart, CHUNK, remaining);
        __builtin_amdgcn_s_wait_tensorcnt(0);
    }
    __syncthreads();

    const int lane = threadIdx.x & 31;           // wave32: lane -> feature pair
    const int wave = threadIdx.x >> 5;           // 8 waves / block
    const int fo   = lane * 2;

    for (int e = wave; e < CHUNK; e += 8) {
        float v = sval[e];                       // LDS broadcast reads
        int   r = srow[e];
        int   c = scol[e];
        // tail zero-fill => v=0, r=c=0 => atomically adds 0.0f: benign
        const float2 xv = *(const float2*)(x + (size_t)c * DIM + fo);
        float* dst = y + (size_t)r * DIM + fo;
        gadd_f32(dst + 0, v * xv.x);
        gadd_f32(dst + 1, v * xv.y);
    }
}

// ---- acc = (acc + add) * scale  (scale=1 for inner layers, 0.25 last) --
__global__ __launch_bounds__(256) void lgcn_accum(const float4* __restrict__ add,
                                                  float4* __restrict__ acc,
                                                  float scale, unsigned n4) {
    unsigned i = blockIdx.x * 256u + threadIdx.x;
    if (i >= n4) return;
    float4 a = acc[i];
    float4 b = add[i];
    a.x = (a.x + b.x) * scale;
    a.y = (a.y + b.y) * scale;
    a.z = (a.z + b.z) * scale;
    a.w = (a.w + b.w) * scale;
    acc[i] = a;
}

// ---- three 4096-row gathers out of the finished final matrix ------------
// One float4 (16B) per thread; 3*4096*16 float4 elements total.
__global__ __launch_bounds__(256) void lgcn_gather(const int* __restrict__ users,
                                                   const int* __restrict__ pos,
                                                   const int* __restrict__ neg,
                                                   const float4* __restrict__ finalAll,
                                                   float4* __restrict__ out) {
    int tid = blockIdx.x * 256 + threadIdx.x;
    if (tid >= 3 * BATCH * 16) return;           // 16 float4 per row
    int set = tid / (BATCH * 16);
    int rem = tid - set * (BATCH * 16);
    int r = rem >> 4;                            // row within batch
    int f = rem & 15;                            // float4 within row
    const float4* base;
    int idx;
    if (set == 0)      { idx = users[r]; base = finalAll; }
    else if (set == 1) { idx = pos[r];   base = finalAll + (size_t)U_NODES * 16; }
    else               { idx = neg[r];   base = finalAll + (size_t)U_NODES * 16; }
    out[tid] = base[(size_t)idx * 16 + f];
}

extern "C" void kernel_launch(void* const* d_in, const int* in_sizes, int n_in,
                              void* d_out, int out_size, void* d_ws, size_t ws_size,
                              hipStream_t stream) {
    const float* user_emb = (const float*)d_in[0];
    const float* item_emb = (const float*)d_in[1];
    const float* adj_val  = (const float*)d_in[2];
    const int*   adj_row  = (const int*)d_in[3];
    const int*   adj_col  = (const int*)d_in[4];
    const int*   users    = (const int*)d_in[5];
    const int*   pos      = (const int*)d_in[6];
    const int*   neg      = (const int*)d_in[7];
    // d_in[8] = num_layers (device scalar; fixed 3, cannot sync-read under capture)
    const int nnz = in_sizes[2];

    float* buf0 = (float*)d_ws;                              // 38.4 MB
    float* buf1 = buf0 + (size_t)N_NODES * DIM;              // 38.4 MB
    float* acc  = (float*)d_out + (size_t)3 * BATCH * DIM;   // = u_g || i_g region

    const unsigned n4 = (unsigned)((size_t)N_NODES * DIM / 4);   // float4 count
    const unsigned ew = (n4 + 255u) / 256u;

    lgcn_init<<<ew, 256, 0, stream>>>((const float4*)user_emb, (const float4*)item_emb,
                                      (float4*)buf0, (float4*)acc, n4);

    float* cur = buf0;
    float* nxt = buf1;
    const int nChunks = (nnz + CHUNK - 1) / CHUNK;
    for (int l = 0; l < 3; ++l) {
        lgcn_zero<<<ew, 256, 0, stream>>>((float4*)nxt, n4);
        lgcn_spmm<<<nChunks, 256, 0, stream>>>(adj_val, adj_row, adj_col, cur, nxt, nnz);
        float sc = (l == 2) ? 0.25f : 1.0f;
        lgcn_accum<<<ew, 256, 0, stream>>>((const float4*)nxt, (float4*)acc, sc, n4);
        float* t = cur; cur = nxt; nxt = t;
    }

    lgcn_gather<<<(3 * BATCH * 16 + 255) / 256, 256, 0, stream>>>(users, pos, neg,
                                                                  (const float4*)acc,
                                                                  (float4*)d_out);
}